// GlobalGuideBlock_61993557950807
// MI455X (gfx1250) — compile-verified
//
#include <hip/hip_runtime.h>

// ---------------------------------------------------------------------------
// GlobalGuideBlock for MI455X (gfx1250, wave32).
// Bandwidth-bound (~196MB fp32 @ 23.3TB/s => ~8.4us floor); fp32 throughout.
// Matrix work: V_WMMA_F32_16X16X4_F32 (K=32 contractions = 8 K-steps).
// Phase-1 tile loads use the Tensor Data Mover (tensor_load_to_lds) with the
// D# pad feature producing the padded stride-34 LDS layout directly.
// LDS layouts are transposed so each WMMA operand fetch is one ds_load_b64
// (compiler fuses pairs into ds_load_2addr_b64).
// ---------------------------------------------------------------------------

typedef float v2f __attribute__((ext_vector_type(2)));
typedef float v8f __attribute__((ext_vector_type(8)));

typedef unsigned int tdm_u32x4 __attribute__((ext_vector_type(4)));
typedef int          tdm_i32x8 __attribute__((ext_vector_type(8)));
typedef int          tdm_i32x4 __attribute__((ext_vector_type(4)));

// TDM builtin exists on both probed toolchains (clang-22: 5 args,
// clang-23: 6 args). Force it on so the tensor path is definitely emitted;
// can be disabled with -DUSE_TDM=0.
#ifndef USE_TDM
# define USE_TDM 1
#endif

#define Bn   4
#define Cn   128
#define Nn   32768
#define NGn  64
#define LN_EPS 1e-5f
#define S1   34    // phase-1 LDS row stride (even => aligned b64 lane fetches)
#define S2   130   // phase-2 LDS row stride (even, conflict-free)

__global__ void zero_ws_kernel(float* ws, int n) {
    int i = blockIdx.x * blockDim.x + threadIdx.x;
    if (i < n) ws[i] = 0.0f;
}

// v_proj[b, c=h*32+d, k] = sum_{n<64} global_fea[b,n,c] * EF1[n,k]  (tiny)
__global__ void __launch_bounds__(256)
vproj_kernel(const float* __restrict__ gf, const float* __restrict__ EF1,
             float* __restrict__ vproj) {
    int b = blockIdx.x;
    for (int idx = threadIdx.x; idx < Cn * 32; idx += 256) {
        int c = idx >> 5, k = idx & 31;
        float s = 0.0f;
        for (int n = 0; n < NGn; ++n)
            s += gf[(b * NGn + n) * Cn + c] * EF1[n * 32 + k];
        vproj[(b * Cn + c) * 32 + k] = s;
    }
}

// k_proj[b, c, p] += sum_{n in chunk} LN(skip)[b,n,c] * EF2[n,p]
// Block = (chunk of 512 tokens, batch). lnT layout: [c][t], stride S1=34
// (TDM pad: 2 dwords after every 32) so A-operand fetch = 1x ds_load_b64.
__global__ void __launch_bounds__(256)
kproj_kernel(const float* __restrict__ skip, const float* __restrict__ EF2,
             const float* __restrict__ ln_w, const float* __restrict__ ln_b,
             float* __restrict__ kproj) {
    __shared__ float lnT[Cn * S1];     // [channel][token], padded stride 34
    __shared__ float ef2T[32 * S1];    // [p][token],       padded stride 34

    const int b = blockIdx.y;
    const int chunk = blockIdx.x;              // 64 chunks * 512 tokens
    const int tid = threadIdx.x;
    const int lane = tid & 31, wave = tid >> 5;
    const int lanelow = lane & 15, half = lane >> 4;
    const float* skb = skip + (size_t)b * Cn * Nn;

    v8f acc0 = {0,0,0,0,0,0,0,0};
    v8f acc1 = {0,0,0,0,0,0,0,0};

#if USE_TDM
    const unsigned lds_base = (unsigned)(size_t)(void*)&lnT[0];
    // D# group1 (constant across tiles): data_size=4B(code 2), pad_enable,
    // pad_interval=32 dwords (code 4), pad_amount=2 dwords (code 1),
    // tensor_dim0=32768, tensor_dim1=128, tile_dim0=32, tile_dim1=128,
    // tensor_dim0_stride=32768.
    tdm_i32x8 g1;
    g1[0] = (int)((2u << 16) | (1u << 20) | (4u << 22) | (1u << 25));
    g1[1] = (int)0x80000000u;            // tensor_dim0[15:0]=0x8000 in [63:48]
    g1[2] = (int)(128u << 16);           // tensor_dim1[15:0]=128 in [95:80]
    g1[3] = (int)(32u << 16);            // tile_dim0=32 in [127:112]
    g1[4] = (int)128u;                   // tile_dim1=128 in [143:128]
    g1[5] = (int)32768u;                 // tensor_dim0_stride[31:0]
    g1[6] = 0;
    g1[7] = 0;
    const tdm_i32x4 gz = {0, 0, 0, 0};
#endif

    for (int tile = 0; tile < 16; ++tile) {
        const int n0 = chunk * 512 + tile * 32;
#if USE_TDM
        if (wave == 0) {
            unsigned long long ga = (unsigned long long)(size_t)(skb + n0);
            tdm_u32x4 g0;
            g0[0] = 1u;                                  // count=1
            g0[1] = lds_base;                            // lds_addr
            g0[2] = (unsigned)(ga & 0xFFFFFFFFu);        // global_addr lo
            g0[3] = (unsigned)((ga >> 32) & 0x01FFFFFFu) // global_addr hi
                    | 0x80000000u;                       // type=2
#if __clang_major__ >= 23
            __builtin_amdgcn_tensor_load_to_lds(g0, g1, gz, gz,
                                                (tdm_i32x8){0,0,0,0,0,0,0,0}, 0);
#else
            __builtin_amdgcn_tensor_load_to_lds(g0, g1, gz, gz, 0);
#endif
        }
#else
        for (int i = 0; i < 16; ++i) {                   // fallback: plain loads
            int idx = tid + i * 256;
            int c = idx >> 5, t = idx & 31;
            lnT[c * S1 + t] = skb[(size_t)c * Nn + n0 + t];
        }
#endif
        // EF2 tile [p][t]
        for (int i = 0; i < 4; ++i) {
            int idx = tid + i * 256;
            int p = idx & 31, t = idx >> 5;
            ef2T[p * S1 + t] = EF2[(size_t)(n0 + t) * 32 + p];
        }
#if USE_TDM
        __builtin_amdgcn_s_wait_tensorcnt(0);
#endif
        __syncthreads();
        // LayerNorm over C per token; wave handles 4 tokens
        for (int j = 0; j < 4; ++j) {
            int t = wave * 4 + j;
            float s = 0.0f, s2 = 0.0f;
            for (int cc = lane; cc < Cn; cc += 32) {
                float x = lnT[cc * S1 + t];
                s += x; s2 += x * x;
            }
            for (int m = 16; m >= 1; m >>= 1) {
                s  += __shfl_xor(s,  m, 32);
                s2 += __shfl_xor(s2, m, 32);
            }
            float mu  = s * (1.0f / 128.0f);
            float var = s2 * (1.0f / 128.0f) - mu * mu;
            float rs  = rsqrtf(var + LN_EPS);
            for (int cc = lane; cc < Cn; cc += 32) {
                float x = lnT[cc * S1 + t];
                lnT[cc * S1 + t] = (x - mu) * rs * ln_w[cc] + ln_b[cc];
            }
        }
        __syncthreads();
        // WMMA: A[m=c][k=t] = lnT[c][t]; B[k=t][n=p] = ef2T[p][t]
        const int mrow = wave * 16 + lanelow;
        for (int kk = 0; kk < 8; ++kk) {
            int kb = kk * 4 + half * 2;
            v2f a  = *(const v2f*)&lnT[mrow * S1 + kb];
            v2f b0 = *(const v2f*)&ef2T[lanelow * S1 + kb];
            v2f b1 = *(const v2f*)&ef2T[(16 + lanelow) * S1 + kb];
            acc0 = __builtin_amdgcn_wmma_f32_16x16x4_f32(false, a, false, b0,
                                                         (short)0, acc0, false, false);
            acc1 = __builtin_amdgcn_wmma_f32_16x16x4_f32(false, a, false, b1,
                                                         (short)0, acc1, false, false);
        }
        __syncthreads();
    }
    // D layout: VGPR r -> M = r (+8 upper half-lanes), N = lane&15
    const int mbase = wave * 16 + half * 8;
    for (int r = 0; r < 8; ++r) {
        int m = mbase + r;
        atomicAdd(&kproj[((b * Cn + m) << 5) + lanelow],      acc0[r]);
        atomicAdd(&kproj[((b * Cn + m) << 5) + 16 + lanelow], acc1[r]);
    }
}

// Per 16-token tile: LN(out) -> q ; logits = q . k_proj * temp ; softmax(32) ;
// x = attn . v_proj ; scatter with the reference's scrambled reshape:
//   dst[b, c'=n&127, n'=k*1024+h*256+(n>>7)] = x[b,h,n,k]
__global__ void __launch_bounds__(256)
attn_kernel(const float* __restrict__ outT, const float* __restrict__ ln_w,
            const float* __restrict__ ln_b, const float* __restrict__ temp3,
            const float* __restrict__ kproj, const float* __restrict__ vproj,
            float* __restrict__ dst) {
    __shared__ float lnT[16 * S2];     // [token][channel]
    __shared__ float attnT[16 * S2];   // [token][h*32+p]
    __shared__ float kpT[32 * S2];     // [p][c=h*32+d]   (transposed k_proj)
    __shared__ float vpT[32 * S2];     // [k][c=h*32+p]   (transposed v_proj)

    const int b = blockIdx.y;
    const int chunk = blockIdx.x;              // 128 chunks * 256 tokens
    const int tid = threadIdx.x;
    const int lane = tid & 31, wave = tid >> 5;
    const int lanelow = lane & 15, half = lane >> 4;
    const int h = wave >> 1, nt = wave & 1;    // wave <-> (head, N-tile)
    const float th = temp3[h];

    for (int i = 0; i < 16; ++i) {
        int idx = tid + i * 256;
        int p = idx & 31, c = idx >> 5;
        kpT[p * S2 + c] = kproj[((b * Cn + c) << 5) + p];
        vpT[p * S2 + c] = vproj[((b * Cn + c) << 5) + p];
    }
    __syncthreads();

    const float* ob = outT + (size_t)b * Cn * Nn;
    float* db = dst + (size_t)b * Cn * Nn;

    for (int tile = 0; tile < 16; ++tile) {
        const int n0 = chunk * 256 + tile * 16;
        for (int i = 0; i < 8; ++i) {
            int idx = tid + i * 256;
            int t = idx & 15, c = idx >> 4;
            lnT[t * S2 + c] = ob[(size_t)c * Nn + n0 + t];
        }
        __syncthreads();
        for (int j = 0; j < 2; ++j) {          // LN: wave handles 2 tokens
            int t = wave * 2 + j;
            float s = 0.0f, s2 = 0.0f;
            for (int cc = lane; cc < Cn; cc += 32) {
                float x = lnT[t * S2 + cc];
                s += x; s2 += x * x;
            }
            for (int m = 16; m >= 1; m >>= 1) {
                s  += __shfl_xor(s,  m, 32);
                s2 += __shfl_xor(s2, m, 32);
            }
            float mu  = s * (1.0f / 128.0f);
            float var = s2 * (1.0f / 128.0f) - mu * mu;
            float rs  = rsqrtf(var + LN_EPS);
            for (int cc = lane; cc < Cn; cc += 32) {
                float x = lnT[t * S2 + cc];
                lnT[t * S2 + cc] = (x - mu) * rs * ln_w[cc] + ln_b[cc];
            }
        }
        __syncthreads();
        {   // logits[t][p] = sum_d q[t][h*32+d] * kproj[h*32+d][p]
            v8f acc = {0,0,0,0,0,0,0,0};
            for (int kk = 0; kk < 8; ++kk) {
                int kb = kk * 4 + half * 2;
                v2f a  = *(const v2f*)&lnT[lanelow * S2 + h * 32 + kb];
                v2f bb = *(const v2f*)&kpT[(nt * 16 + lanelow) * S2 + h * 32 + kb];
                acc = __builtin_amdgcn_wmma_f32_16x16x4_f32(false, a, false, bb,
                                                            (short)0, acc, false, false);
            }
            for (int r = 0; r < 8; ++r) {
                int t = r + half * 8;
                attnT[t * S2 + h * 32 + nt * 16 + lanelow] = acc[r] * th;
            }
        }
        __syncthreads();
        if (tid < 64) {                        // softmax over 32, rows (t, head)
            int t = tid >> 2, hh = tid & 3;
            float* row = &attnT[t * S2 + hh * 32];
            float mx = row[0];
            for (int p = 1; p < 32; ++p) mx = fmaxf(mx, row[p]);
            float s = 0.0f;
            for (int p = 0; p < 32; ++p) { float e = __expf(row[p] - mx); row[p] = e; s += e; }
            float inv = 1.0f / s;
            for (int p = 0; p < 32; ++p) row[p] *= inv;
        }
        __syncthreads();
        {   // x[t][k] = sum_p attn[t][h*32+p] * vproj[h*32+p][k] ; scatter
            v8f acc = {0,0,0,0,0,0,0,0};
            for (int kk = 0; kk < 8; ++kk) {
                int kb = kk * 4 + half * 2;
                v2f a  = *(const v2f*)&attnT[lanelow * S2 + h * 32 + kb];
                v2f bb = *(const v2f*)&vpT[(nt * 16 + lanelow) * S2 + h * 32 + kb];
                acc = __builtin_amdgcn_wmma_f32_16x16x4_f32(false, a, false, bb,
                                                            (short)0, acc, false, false);
            }
            const int kcol = nt * 16 + lanelow;
            for (int r = 0; r < 8; ++r) {
                int t = r + half * 8;
                int nn = n0 + t;
                int cp = nn & 127;
                int npr = (kcol << 10) + (h << 8) + (nn >> 7);
                db[(size_t)cp * Nn + npr] = acc[r];
            }
        }
        __syncthreads();
    }
}

extern "C" void kernel_launch(void* const* d_in, const int* in_sizes, int n_in,
                              void* d_out, int out_size, void* d_ws, size_t ws_size,
                              hipStream_t stream) {
    const float* skip = (const float*)d_in[0];
    const float* outT = (const float*)d_in[1];
    const float* gf   = (const float*)d_in[2];
    const float* ln_w = (const float*)d_in[3];
    const float* ln_b = (const float*)d_in[4];
    const float* temp = (const float*)d_in[5];
    const float* EF1  = (const float*)d_in[6];
    const float* EF2  = (const float*)d_in[7];
    float* dst   = (float*)d_out;
    float* kproj = (float*)d_ws;          // Bn*Cn*32 = 16384 floats
    float* vproj = kproj + Bn * Cn * 32;  // Bn*Cn*32 = 16384 floats

    zero_ws_kernel<<<64, 256, 0, stream>>>(kproj, Bn * Cn * 32);
    vproj_kernel<<<Bn, 256, 0, stream>>>(gf, EF1, vproj);
    kproj_kernel<<<dim3(64, Bn), 256, 0, stream>>>(skip, EF2, ln_w, ln_b, kproj);
    attn_kernel<<<dim3(128, Bn), 256, 0, stream>>>(outT, ln_w, ln_b, temp,
                                                   kproj, vproj, dst);
}